// MixGCF_39470749450557
// MI455X (gfx1250) — compile-verified
//
#include <hip/hip_runtime.h>
#include <math.h>

// MixGCF negative-sampling + BPR loss for gfx1250 (MI455X, wave32).
// Gather-bandwidth bound (~268 MB of candidate rows); scores computed on the
// matrix pipe with V_WMMA_F32_16X16X4_F32 at full f32 precision.
// B-operands live in registers (k-invariant), A-operands are direct
// global->VGPR gathers hidden by global_prefetch; 4 WMMA accumulator chains.
// Row pointers are re-derived from the kernel argument inside the unrolled
// k-loop so the gathers lower to GLOBAL (not FLAT) loads.

typedef __attribute__((ext_vector_type(2))) float v2f;
typedef __attribute__((ext_vector_type(8))) float v8f;

#define HOPS 4
#define DIM  64
#define ROW  (HOPS * DIM)   // 256 floats per embedding row
#define NNEG 64
#define KK   2

__device__ __forceinline__ float wave_sum(float v) {
#pragma unroll
    for (int m = 16; m > 0; m >>= 1) v += __shfl_xor(v, m, 32);
    return v;
}

__global__ __launch_bounds__(128) void mixgcf_main(
    const float* __restrict__ userEmb,   // (N_USERS, 4, 64)
    const float* __restrict__ itemEmb,   // (N_ITEMS, 4, 64)
    const float* __restrict__ seedEmb,   // (B, 1)
    const int*   __restrict__ userIdx,   // (B,)
    const int*   __restrict__ posIdx,    // (B,)
    const int*   __restrict__ negIdx,    // (B, 128)
    float*       __restrict__ ws)        // (B, 2): per-b {mf_b, reg_b}
{
    __shared__ __align__(16) float sE[ROW];        // s_e  (hop-major)
    __shared__ __align__(16) float pE[ROW];        // p_e
    __shared__ __align__(16) float negE[KK * DIM]; // hop-mean of selected negs
    __shared__ float spL[HOPS];                    // <s_e[h], p_e[h]>
    __shared__ float wbest[4][4];                  // per-wave per-hop best score
    __shared__ int   wbn[4][4];                    // per-wave per-hop best slot
    __shared__ int   chosenItem[HOPS];             // selected item ids (per k)
    __shared__ float wpart[2][4];                  // final reduction partials

    const int tid  = threadIdx.x;
    const int lane = tid & 31;
    const int wv   = tid >> 5;       // wave 0..3 -> candidate tile
    const int half = lane >> 4;      // 0 / 1
    const int j    = lane & 15;      // WMMA column index / A row index
    const int cp   = half * 2;       // lane's K sub-pair within each K=4 step

    const int b   = blockIdx.x;
    const int uid = userIdx[b];
    const int pid = posIdx[b];
    const float seed  = seedEmb[b];
    const float seed1 = 1.0f - seed;

    // ---- stage s_e, p_e into LDS (coalesced, 2 elems/thread) ----
    {
        const float* ur = userEmb + (size_t)uid * ROW;
        const float* pr = itemEmb + (size_t)pid * ROW;
        sE[tid]       = ur[tid];
        sE[tid + 128] = ur[tid + 128];
        pE[tid]       = pr[tid];
        pE[tid + 128] = pr[tid + 128];
    }
    __syncthreads();

    // ---- sp[h] = <s_e[h], p_e[h]> : wave w handles hop w ----
    {
        float v = sE[wv * DIM + lane]      * pE[wv * DIM + lane]
                + sE[wv * DIM + lane + 32] * pE[wv * DIM + lane + 32];
        v = wave_sum(v);
        if (lane == 0) spL[wv] = v;
    }

    // ---- candidate ids for both k rounds; prefetch all rows up front ----
    // Keep ids as ints (not pointers) so later address computation stays
    // derived from the global kernel argument -> global_load, not flat_load.
    const int slot = wv * 16 + j;                              // n in [0,64)
    int itemIdK[KK];
#pragma unroll
    for (int k = 0; k < KK; ++k) {
        itemIdK[k] = negIdx[(size_t)b * (KK * NNEG) + k * NNEG + slot];
        const float* rb = itemEmb + (size_t)itemIdK[k] * ROW;
#pragma unroll
        for (int pf = 0; pf < ROW; pf += 64)
            __builtin_prefetch(rb + pf, 0, 1);
    }
    __syncthreads();   // sE/pE/spL visible

    // ---- B operands in registers: lane j's only nonzero hop is h == j ----
    // (lanes j>=4 never match any h; their Breg content is masked to zero
    //  in the inner loop, so load from hop j&3 unpredicated)
    v2f Breg[16];
    {
        const float* sb = &sE[(j & 3) * DIM];
#pragma unroll
        for (int t = 0; t < 16; ++t)
            Breg[t] = *(const v2f*)(sb + 4 * t + cp);
    }

    // ---- K = 2 rounds of negative sampling (fully unrolled) ----
#pragma unroll
    for (int k = 0; k < KK; ++k) {
        const float* rowbase = itemEmb + (size_t)itemIdK[k] * ROW;

        // raw[n,h] via chained f32 WMMA: K runs over flattened (hop,channel),
        // B column j is nonzero only on hop j -> D column h = raw[.,h].
        // 4 accumulator chains keep the matrix pipe issuing back-to-back.
        v8f acc[4];
#pragma unroll
        for (int q = 0; q < 4; ++q) acc[q] = (v8f){};
#pragma unroll
        for (int h = 0; h < HOPS; ++h) {
            const float* hb = rowbase + h * DIM;
            const bool mine = (j == h);
#pragma unroll
            for (int t = 0; t < 16; ++t) {
                v2f a  = *(const v2f*)(hb + 4 * t + cp);
                v2f bb = mine ? Breg[t] : (v2f){0.0f, 0.0f};
                acc[t & 3] = __builtin_amdgcn_wmma_f32_16x16x4_f32(
                    false, a, false, bb, (short)0, acc[t & 3], false, false);
            }
        }

        // D layout: VGPR r holds rows {r, r+8}; this lane sees column j,
        // rows half*8 + r.  Column j==h carries raw[.,h].
        float best  = -INFINITY;
        int   bestn = 0;
        if (j < HOPS) {
            const float sp_j = spL[j];
#pragma unroll
            for (int r = 0; r < 8; ++r) {
                const float raw = (acc[0][r] + acc[1][r]) + (acc[2][r] + acc[3][r]);
                const float sc  = seed * sp_j + seed1 * raw;
                const int   n   = wv * 16 + half * 8 + r;   // ascending in r
                if (sc > best) { best = sc; bestn = n; }    // first-occurrence
            }
        }
        // merge the two row-halves (lane j <-> lane j+16); lower n wins ties
        {
            const float ob = __shfl_xor(best, 16, 32);
            const int   on = __shfl_xor(bestn, 16, 32);
            if (ob > best || (ob == best && on < bestn)) { best = ob; bestn = on; }
        }
        if (half == 0 && j < HOPS) { wbest[wv][j] = best; wbn[wv][j] = bestn; }
        __syncthreads();

        // merge the 4 waves (tiles are in ascending-n order)
        if (tid < HOPS) {
            float bb = wbest[0][tid];
            int   nn = wbn[0][tid];
#pragma unroll
            for (int w = 1; w < 4; ++w) {
                if (wbest[w][tid] > bb) { bb = wbest[w][tid]; nn = wbn[w][tid]; }
            }
            chosenItem[tid] = negIdx[(size_t)b * (KK * NNEG) + k * NNEG + nn];
        }
        __syncthreads();

        // reconstruct selected mixed embeddings, hop-mean -> negE[k]
        if (tid < DIM) {
            const int c = tid;
            float accv = 0.0f;
#pragma unroll
            for (int h = 0; h < HOPS; ++h) {
                const float v  = itemEmb[(size_t)chosenItem[h] * ROW + h * DIM + c];
                const float ne = seed * pE[h * DIM + c] + seed1 * v;
                accv += ne;
            }
            negE[k * DIM + c] = accv * 0.25f;
        }
    }
    __syncthreads();

    // ---- BPR loss partials for this batch element ----
    if (tid < DIM) {
        const int c = tid;
        const float u  = 0.25f * (sE[c] + sE[DIM + c] + sE[2 * DIM + c] + sE[3 * DIM + c]);
        const float pe = 0.25f * (pE[c] + pE[DIM + c] + pE[2 * DIM + c] + pE[3 * DIM + c]);
        const float n0 = negE[c];
        const float n1 = negE[DIM + c];
        float v0 = u * pe;                                   // pos score
        float v1 = u * n0;                                   // neg score k=0
        float v2 = u * n1;                                   // neg score k=1
        float v3 = u * u + pe * pe + n0 * n0 + n1 * n1;      // regularizer
        v0 = wave_sum(v0); v1 = wave_sum(v1); v2 = wave_sum(v2); v3 = wave_sum(v3);
        if (lane == 0) {
            wpart[wv][0] = v0; wpart[wv][1] = v1;
            wpart[wv][2] = v2; wpart[wv][3] = v3;
        }
    }
    __syncthreads();
    if (tid == 0) {
        const float ps  = wpart[0][0] + wpart[1][0];
        const float ns0 = wpart[0][1] + wpart[1][1];
        const float ns1 = wpart[0][2] + wpart[1][2];
        const float rg  = wpart[0][3] + wpart[1][3];
        const float mf  = log1pf(expf(ns0 - ps) + expf(ns1 - ps));
        ws[2 * (size_t)b]     = mf;
        ws[2 * (size_t)b + 1] = rg;
    }
}

// Deterministic final reduction (fixed strided + tree order; replay-safe).
__global__ __launch_bounds__(256) void mixgcf_reduce(
    const float* __restrict__ ws, float* __restrict__ out, int B)
{
    __shared__ float smf[256];
    __shared__ float srg[256];
    const int t = threadIdx.x;
    float mf = 0.0f, rg = 0.0f;
    for (int i = t; i < B; i += 256) {
        mf += ws[2 * (size_t)i];
        rg += ws[2 * (size_t)i + 1];
    }
    smf[t] = mf; srg[t] = rg;
    __syncthreads();
    for (int s = 128; s > 0; s >>= 1) {
        if (t < s) { smf[t] += smf[t + s]; srg[t] += srg[t + s]; }
        __syncthreads();
    }
    if (t == 0) {
        const float mf_loss  = smf[0] / (float)B;
        const float emb_loss = 0.0001f * (0.5f * srg[0]) / (float)B;
        out[0] = mf_loss + emb_loss;
        out[1] = mf_loss;
        out[2] = emb_loss;
    }
}

extern "C" void kernel_launch(void* const* d_in, const int* in_sizes, int n_in,
                              void* d_out, int out_size, void* d_ws, size_t ws_size,
                              hipStream_t stream) {
    const float* userEmb = (const float*)d_in[0];
    const float* itemEmb = (const float*)d_in[1];
    const float* seedEmb = (const float*)d_in[2];
    const int*   userIdx = (const int*)d_in[3];
    const int*   posIdx  = (const int*)d_in[4];
    const int*   negIdx  = (const int*)d_in[5];
    const int B = in_sizes[3];          // 2048

    float* ws = (float*)d_ws;           // needs 2*B floats (16 KB)

    mixgcf_main<<<B, 128, 0, stream>>>(userEmb, itemEmb, seedEmb,
                                       userIdx, posIdx, negIdx, ws);
    mixgcf_reduce<<<1, 256, 0, stream>>>(ws, (float*)d_out, B);
}